// QACModule_27161373180353
// MI455X (gfx1250) — compile-verified
//
#include <hip/hip_runtime.h>
#include <hip/hip_bf16.h>

// Problem constants (match reference)
#define B_ 8
#define S_ 8192
#define D_ 64
#define DM 1024
#define EPS_ 1e-8f

#define ROWS (B_ * S_)          // 65536
#define KC 32                   // K chunk staged in LDS
// LDS layout for W2 chunk: element (k,n) at (k>>1)*PSTRIDE + 2n + (k&1).
// PSTRIDE % 64 == 32 -> the two half-waves (pair-rows 1 apart) hit disjoint
// 32-bank ranges on every ds_load_b64 -> conflict-free.
#define PSTRIDE 288             // floats per k-pair row (2*128 data + 32 pad)

typedef float v2f __attribute__((ext_vector_type(2)));
typedef float v8f __attribute__((ext_vector_type(8)));

// ---------------------------------------------------------------------------
// Kernel 1: per-row scalar xi.  One wave32 per row, lane i holds elements
// i and i+32 of the 64-wide state vectors.  xi written to d_out tail (the
// reference's second output) and consumed by the GEMM kernel.
// ---------------------------------------------------------------------------
__global__ __launch_bounds__(256) void qac_xi_kernel(
    const float* __restrict__ pq, const float* __restrict__ pk,
    const float* __restrict__ pv, float* __restrict__ xi_out) {
  const int lane = threadIdx.x & 31;
  const int wave = threadIdx.x >> 5;
  const int row  = blockIdx.x * 8 + wave;
  const size_t base = (size_t)row * D_;

  float q0 = pq[base + lane], q1 = pq[base + lane + 32];
  float k0 = pk[base + lane], k1 = pk[base + lane + 32];
  float v0 = pv[base + lane], v1 = pv[base + lane + 32];

  float sq   = q0 + q1;            // sum(psi_Q)
  float sq2  = q0 * q0 + q1 * q1;  // ||psi_Q||^2
  float sk2  = k0 * k0 + k1 * k1;  // ||psi_K||^2
  float svlo = v0 * v0;            // sum psi_V^2, idx < 32
  float svhi = v1 * v1;            // idx >= 32

#pragma unroll
  for (int m = 16; m >= 1; m >>= 1) {
    sq   += __shfl_xor(sq, m, 32);
    sq2  += __shfl_xor(sq2, m, 32);
    sk2  += __shfl_xor(sk2, m, 32);
    svlo += __shfl_xor(svlo, m, 32);
    svhi += __shfl_xor(svhi, m, 32);
  }
  float kfirst = __shfl(k0, 0, 32);  // psi_K[row, 0]

  if (lane == 0) {
    float nq = sqrtf(sq2) + EPS_;
    float nk = sqrtf(sk2) + EPS_;
    float nv = sqrtf(svlo + svhi) + EPS_;
    float sumq = sq / nq;            // sum of normalized q
    float kh   = kfirst / nk;        // normalized k[0]
    float att  = sumq * sumq * kh * kh * (1.0f / (float)D_);
    float vexp = (svlo - svhi) / (nv * nv);
    xi_out[row] = att * vexp;
  }
}

// ---------------------------------------------------------------------------
// Kernel 2: yQ = relu(xi (x) W1 + b1) @ W2 + b2 as a fused WMMA fp32 GEMM.
// A-matrix (h) is synthesized in registers from one scalar per row plus the
// LDS-staged W1/b1 chunk (separate arrays -> one ds_load_2addr_b64 per A
// fragment); W2 chunk sits in LDS k-pair-interleaved so every B fragment is
// a single aligned ds_load_b64 (pairs merged into ds_load_2addr_b64).
// Block = 256 threads (8 waves), block tile 128x128, wave tile 16x128.
// ---------------------------------------------------------------------------
__global__ __launch_bounds__(256) void qac_gemm_kernel(
    const float* __restrict__ xi, const float* __restrict__ W1,
    const float* __restrict__ b1, const float* __restrict__ W2,
    const float* __restrict__ b2, float* __restrict__ out) {
  __shared__ float Bs[(KC / 2) * PSTRIDE];  // 18 KB, k-pair interleaved
  __shared__ float W1s[KC];                 // W1 chunk (pair-contiguous)
  __shared__ float b1s[KC];                 // b1 chunk (pair-contiguous)

  const int lane = threadIdx.x & 31;
  const int wave = threadIdx.x >> 5;
  const int hf   = lane >> 4;    // half-wave: selects k pair within K=4 step
  const int ln   = lane & 15;    // m index (A) / n index (B,C,D)
  const int mb   = blockIdx.x * 128;
  const int nb   = blockIdx.y * 128;
  const int rowBase = mb + wave * 16;

  // Staging coordinates: pair-row p, 16 threads per pair-row.
  const int sp = threadIdx.x >> 4;   // 0..15 : k-pair row
  const int sq = threadIdx.x & 15;   // 0..15 : float4 slot phase

  // Row scalar for this wave's 16 rows (lanes 16-31 duplicate lanes 0-15).
  const float xim = xi[rowBase + ln];

  v8f acc[8] = {};  // 8 tiles of 16x16 fp32 accumulators

  for (int k0 = 0; k0 < DM; k0 += KC) {
    // ---- stage W2[k0:k0+32, nb:nb+128] into LDS, k-pair interleaved ----
    {
      const float* r0 = W2 + (size_t)(k0 + 2 * sp) * DM + nb;
      const float* r1 = r0 + DM;
#pragma unroll
      for (int j = 0; j < 4; ++j) {
        const int s   = sq + 16 * j;  // float4 slot: covers cols 2s, 2s+1
        const int col = 2 * s;
        float2 e = *(const float2*)(r0 + col);  // k even: W2[2p, col..col+1]
        float2 o = *(const float2*)(r1 + col);  // k odd : W2[2p+1, col..col+1]
        float4 st = make_float4(e.x, o.x, e.y, o.y);
        *(float4*)&Bs[sp * PSTRIDE + 4 * s] = st;
      }
      // Pull next chunk toward L2 while we compute on this one.
      if (k0 + KC < DM) {
        __builtin_prefetch(W2 + (size_t)(k0 + KC + 2 * sp) * DM + nb + 2 * sq, 0, 1);
      }
    }
    if (threadIdx.x < KC) {
      W1s[threadIdx.x] = W1[k0 + threadIdx.x];
    } else if (threadIdx.x < 2 * KC) {
      b1s[threadIdx.x - KC] = b1[k0 + threadIdx.x - KC];
    }
    __syncthreads();

    // ---- 8 WMMA K-steps of 4 over this chunk ----
#pragma unroll
    for (int kk = 0; kk < KC; kk += 4) {
      const int ka = kk + 2 * hf;  // this half-wave's k pair {ka, ka+1}
      // Two aligned b64 pair loads (compiler fuses into ds_load_2addr_b64):
      v2f w  = *(const v2f*)&W1s[ka];   // {W1[ka], W1[ka+1]}
      v2f bi = *(const v2f*)&b1s[ka];   // {b1[ka], b1[ka+1]}
      // A fragment (16x4 fp32): lane holds h[m=ln, ka], h[m=ln, ka+1]
      v2f a;
      a.x = fmaxf(fmaf(xim, w.x, bi.x), 0.0f);
      a.y = fmaxf(fmaf(xim, w.y, bi.y), 0.0f);
      const float* bp = &Bs[(ka >> 1) * PSTRIDE];
#pragma unroll
      for (int t = 0; t < 8; ++t) {
        // B fragment: one aligned b64 = {W2[ka, n], W2[ka+1, n]}
        v2f b = *(const v2f*)(bp + 2 * (16 * t + ln));
        acc[t] = __builtin_amdgcn_wmma_f32_16x16x4_f32(
            /*neg_a=*/false, a, /*neg_b=*/false, b,
            /*c_mod=*/(short)0, acc[t], /*reuse_a=*/false, /*reuse_b=*/false);
      }
    }
    __syncthreads();
  }

  // ---- epilogue: + b2, store per C/D layout (VGPR r -> rows r, r+8) ----
#pragma unroll
  for (int t = 0; t < 8; ++t) {
    const int col = nb + 16 * t + ln;
    const float bias = b2[col];
#pragma unroll
    for (int r = 0; r < 8; ++r) {
      const int row = rowBase + r + 8 * hf;
      out[(size_t)row * DM + col] = acc[t][r] + bias;
    }
  }
}

// ---------------------------------------------------------------------------
extern "C" void kernel_launch(void* const* d_in, const int* in_sizes, int n_in,
                              void* d_out, int out_size, void* d_ws, size_t ws_size,
                              hipStream_t stream) {
  const float* psi_Q = (const float*)d_in[0];
  const float* psi_K = (const float*)d_in[1];
  const float* psi_V = (const float*)d_in[2];
  const float* W1    = (const float*)d_in[3];
  const float* b1    = (const float*)d_in[4];
  const float* W2    = (const float*)d_in[5];
  const float* b2    = (const float*)d_in[6];

  float* yQ = (float*)d_out;
  float* xi = yQ + (size_t)ROWS * DM;  // second reference output, also GEMM input

  // 65536 rows, 8 rows (waves) per block
  qac_xi_kernel<<<ROWS / 8, 256, 0, stream>>>(psi_Q, psi_K, psi_V, xi);

  // 128x128 block tiles over [65536 x 1024]
  dim3 grid(ROWS / 128, DM / 128);
  qac_gemm_kernel<<<grid, 256, 0, stream>>>(xi, W1, b1, W2, b2, yQ);
}